// RevEncoder_52621939310803
// MI455X (gfx1250) — compile-verified
//
#include <hip/hip_runtime.h>
#include <hip/hip_bf16.h>
#include <math.h>

typedef float v2f __attribute__((ext_vector_type(2)));
typedef float v8f __attribute__((ext_vector_type(8)));

#define T_LEN   256
#define BATCH   64
#define HID     512
#define HALF    256
#define KDIM    768
#define KQ      (KDIM / 4)      // 192 WMMA K-steps
#define KQX     (HID / 4)       // 128 K-steps in the x part
#define SLICE_  100
#define RING    4
#define NWAVE   16
#define MAXF    0.875f
#define STRIP_F (KQ * 64)       // floats per packed 16-column weight strip (12288)

struct Params {
  const int*   seq;
  const float* emb;
  const float* PW[2][4];  // packed Wzr1, Wg1, Wzr2, Wg2 per layer (in ws)
  const float* B[2][4];   // bzr1, bg1, bzr2, bg2 per layer
  float* out;
  float* ring;
  int*   flags;           // flags[0]=steps produced by layer0, flags[1]=consumed by layer1
};

#define WMMA_F32(a, b, c) \
  __builtin_amdgcn_wmma_f32_16x16x4_f32(false, (a), false, (b), (short)0, (c), false, false)

// --- K=768 fp32 WMMA reduction: NS N-strips x 4 M-tiles per wave -----------
// B comes from the packed layout (one coalesced b64 per strip per K-step);
// A is loaded once and shared across strips: 0.75 loads/WMMA at NS=2.
// A (16x4 f32): lanes0-15 hold M rows K={k,k+1}; lanes16-31 K={k+2,k+3}.
template <int NS>
__device__ __forceinline__ void gemm_strips(const float* __restrict__ xbase,
                                            const int*   __restrict__ rowoff,
                                            const float* __restrict__ hpart,
                                            const float* const Bp[NS],
                                            int lane, v8f acc[NS][4]) {
  const int ml = lane & 15;
  const int kh = (lane >> 4) * 2;
  const float* xr0 = xbase + rowoff[ml];
  const float* xr1 = xbase + rowoff[16 + ml];
  const float* xr2 = xbase + rowoff[32 + ml];
  const float* xr3 = xbase + rowoff[48 + ml];
  const float* hr0 = hpart + (ml)      * HALF;
  const float* hr1 = hpart + (16 + ml) * HALF;
  const float* hr2 = hpart + (32 + ml) * HALF;
  const float* hr3 = hpart + (48 + ml) * HALF;
#pragma unroll 4
  for (int kq = 0; kq < KQX; ++kq) {           // x part of the K dimension
    const int kb = kq * 4 + kh;
    v2f b[NS];
#pragma unroll
    for (int s = 0; s < NS; ++s)
      b[s] = *reinterpret_cast<const v2f*>(Bp[s] + (size_t)kq * 64 + lane * 2);
    v2f a0 = *reinterpret_cast<const v2f*>(xr0 + kb);
    v2f a1 = *reinterpret_cast<const v2f*>(xr1 + kb);
    v2f a2 = *reinterpret_cast<const v2f*>(xr2 + kb);
    v2f a3 = *reinterpret_cast<const v2f*>(xr3 + kb);
#pragma unroll
    for (int s = 0; s < NS; ++s) {
      acc[s][0] = WMMA_F32(a0, b[s], acc[s][0]);
      acc[s][1] = WMMA_F32(a1, b[s], acc[s][1]);
      acc[s][2] = WMMA_F32(a2, b[s], acc[s][2]);
      acc[s][3] = WMMA_F32(a3, b[s], acc[s][3]);
    }
  }
#pragma unroll 4
  for (int kq = KQX; kq < KQ; ++kq) {          // h / (r*h) part from LDS
    const int kb = (kq - KQX) * 4 + kh;
    v2f b[NS];
#pragma unroll
    for (int s = 0; s < NS; ++s)
      b[s] = *reinterpret_cast<const v2f*>(Bp[s] + (size_t)kq * 64 + lane * 2);
    v2f a0 = *reinterpret_cast<const v2f*>(hr0 + kb);
    v2f a1 = *reinterpret_cast<const v2f*>(hr1 + kb);
    v2f a2 = *reinterpret_cast<const v2f*>(hr2 + kb);
    v2f a3 = *reinterpret_cast<const v2f*>(hr3 + kb);
#pragma unroll
    for (int s = 0; s < NS; ++s) {
      acc[s][0] = WMMA_F32(a0, b[s], acc[s][0]);
      acc[s][1] = WMMA_F32(a1, b[s], acc[s][1]);
      acc[s][2] = WMMA_F32(a2, b[s], acc[s][2]);
      acc[s][3] = WMMA_F32(a3, b[s], acc[s][3]);
    }
  }
}

// zr GEMM: [x, hpart] @ Wzr + b -> sigmoid.
// Strip 0 (n0 = wv*16 < 256)  -> zbuf = 0.875*s + 0.125
// Strip 1 (n0 = wv*16 + 256)  -> rhbuf = r * hpart
__device__ __forceinline__ void zr_pass(const float* xbase, const int* rowoff,
                                        const float* hpart, const float* PW,
                                        const float* bias, float* zbuf, float* rhbuf,
                                        int wv, int lane) {
  v8f acc[2][4] = {};
  const float* Bp[2] = { PW + (size_t)wv * STRIP_F,
                         PW + (size_t)(wv + 16) * STRIP_F };
  gemm_strips<2>(xbase, rowoff, hpart, Bp, lane, acc);
  const int ml = lane & 15;
  const int mh = (lane >> 4) * 8;
  const int ncz = wv * 16 + ml;          // z column (0..255)
  const float bz = bias[ncz];
  const float br = bias[ncz + HALF];     // r column (256..511)
#pragma unroll
  for (int j = 0; j < 4; ++j) {
#pragma unroll
    for (int i = 0; i < 8; ++i) {
      const int   m  = j * 16 + i + mh;
      const float sz = 1.f / (1.f + __expf(-(acc[0][j][i] + bz)));
      const float sr = 1.f / (1.f + __expf(-(acc[1][j][i] + br)));
      zbuf[m * HALF + ncz]  = MAXF * sz + (1.f - MAXF);
      rhbuf[m * HALF + ncz] = sr * hpart[m * HALF + ncz];
    }
  }
}

// g GEMM: [x, rhbuf] @ Wg + b -> tanh; quantized half-update of hdst in LDS
__device__ __forceinline__ void g_pass(const float* xbase, const int* rowoff,
                                       const float* rhbuf, const float* PW,
                                       const float* bias, const float* zbuf,
                                       float* hdst, int wv, int lane) {
  v8f acc[1][4] = {};
  const float* Bp[1] = { PW + (size_t)wv * STRIP_F };
  gemm_strips<1>(xbase, rowoff, rhbuf, Bp, lane, acc);
  const int   nc = wv * 16 + (lane & 15);
  const float bb = bias[nc];
  const int   mh = (lane >> 4) * 8;
#pragma unroll
  for (int j = 0; j < 4; ++j) {
#pragma unroll
    for (int i = 0; i < 8; ++i) {
      const int   m  = j * 16 + i + mh;
      const float g  = tanhf(acc[0][j][i] + bb);
      const float z  = zbuf[m * HALF + nc];
      const float h0 = hdst[m * HALF + nc];
      const float zq = floorf(z * 1024.f) * (1.f / 1024.f);               // qfloor FR=10
      float h = floorf(h0 * zq * 8388608.f) * (1.f / 8388608.f);          // qfloor HR=23
      h += rintf((1.f - z) * g * 8388608.f) * (1.f / 8388608.f);          // qround HR=23
      hdst[m * HALF + nc] = h;
    }
  }
}

// One-time (per launch) weight repack into per-lane WMMA B layout:
// dst[nt*KQ*64 + kq*64 + lane*2 + {0,1}] = W[(4kq + (lane>>4)*2 + {0,1})][nt*16 + lane&15]
__global__ void __launch_bounds__(256) repack_kernel(const float* __restrict__ W,
                                                     float* __restrict__ dst, int N) {
  const int idx = blockIdx.x * blockDim.x + threadIdx.x;
  const int total = (N / 16) * KQ * 32;
  if (idx >= total) return;
  const int L  = idx & 31;
  const int kq = (idx >> 5) % KQ;
  const int nt = idx / (KQ * 32);
  const int n  = nt * 16 + (L & 15);
  const int k  = kq * 4 + (L >> 4) * 2;
  dst[(size_t)idx * 2]     = W[(size_t)k * N + n];
  dst[(size_t)idx * 2 + 1] = W[(size_t)(k + 1) * N + n];
}

__global__ void __launch_bounds__(512) init_kernel(float* out, int* flags) {
  const int i = blockIdx.x * blockDim.x + threadIdx.x;
  if (i < 2) flags[i] = 0;
  float* sv0 = out + 2 * BATCH * HID;   // saved[0] = zeros
  if (i < BATCH * SLICE_) sv0[i] = 0.f;
}

__global__ void __launch_bounds__(512) rnn_kernel(Params p) {
  extern __shared__ char smem_raw[];
  float* h1     = reinterpret_cast<float*>(smem_raw);
  float* h2     = h1 + BATCH * HALF;
  float* zbuf   = h2 + BATCH * HALF;
  float* rhbuf  = zbuf + BATCH * HALF;
  int*   rowoff = reinterpret_cast<int*>(rhbuf + BATCH * HALF);

  const int layer = blockIdx.x;
  const int tid   = threadIdx.x;
  const int wv    = tid >> 5;
  const int lane  = tid & 31;

  for (int i = tid; i < BATCH * HALF; i += 512) { h1[i] = 0.f; h2[i] = 0.f; }
  __syncthreads();

  const float* xbase;
  for (int t = 0; t < T_LEN; ++t) {
    // -- acquire this step's input -----------------------------------------
    if (layer == 0) {
      if (tid == 0) {   // wait until ring slot is free
        while (__hip_atomic_load(&p.flags[1], __ATOMIC_ACQUIRE,
                                 __HIP_MEMORY_SCOPE_AGENT) + RING <= t)
          __builtin_amdgcn_s_sleep(2);
      }
      __syncthreads();
      if (tid < BATCH) rowoff[tid] = p.seq[t * BATCH + tid] * HID;  // gather rows of emb
      xbase = p.emb;
    } else {
      if (tid == 0) {   // wait until layer0 finished step t
        while (__hip_atomic_load(&p.flags[0], __ATOMIC_ACQUIRE,
                                 __HIP_MEMORY_SCOPE_AGENT) <= t)
          __builtin_amdgcn_s_sleep(2);
      }
      __syncthreads();
      if (tid < BATCH) rowoff[tid] = tid * HID;
      xbase = p.ring + (size_t)(t & (RING - 1)) * BATCH * HID;
    }
    __syncthreads();

    // -- sub-cell 1: zr1 = sig([x,h2]Wzr1), g1, update h1 ------------------
    zr_pass(xbase, rowoff, h2, p.PW[layer][0], p.B[layer][0], zbuf, rhbuf, wv, lane);
    __syncthreads();
    g_pass(xbase, rowoff, rhbuf, p.PW[layer][1], p.B[layer][1], zbuf, h1, wv, lane);
    __syncthreads();
    // -- sub-cell 2: zr2 = sig([x,h1]Wzr2), g2, update h2 ------------------
    zr_pass(xbase, rowoff, h1, p.PW[layer][2], p.B[layer][2], zbuf, rhbuf, wv, lane);
    __syncthreads();
    g_pass(xbase, rowoff, rhbuf, p.PW[layer][3], p.B[layer][3], zbuf, h2, wv, lane);
    __syncthreads();

    // -- hand-off / outputs ------------------------------------------------
    if (layer == 0) {
      float* dst = p.ring + (size_t)(t & (RING - 1)) * BATCH * HID;
      for (int i = tid; i < BATCH * HALF; i += 512) {
        const int m = i >> 8, c = i & 255;
        dst[m * HID + c]        = h1[i];
        dst[m * HID + HALF + c] = h2[i];
      }
      __threadfence();
      __syncthreads();
      if (tid == 0)
        __hip_atomic_store(&p.flags[0], t + 1, __ATOMIC_RELEASE,
                           __HIP_MEMORY_SCOPE_AGENT);
    } else {
      float* sv = p.out + 2 * BATCH * HID + (size_t)(t + 1) * BATCH * SLICE_;
      for (int i = tid; i < BATCH * SLICE_; i += 512) {
        const int m = i / SLICE_, c = i - m * SLICE_;
        sv[i] = h1[m * HALF + c];     // first-half columns 0..99
      }
      __threadfence();
      __syncthreads();
      if (tid == 0)
        __hip_atomic_store(&p.flags[1], t + 1, __ATOMIC_RELEASE,
                           __HIP_MEMORY_SCOPE_AGENT);
    }
  }

  // -- final carry: out[layer][b][:] = [h1|h2] -----------------------------
  float* dst = p.out + (size_t)layer * BATCH * HID;
  for (int i = tid; i < BATCH * HALF; i += 512) {
    const int m = i >> 8, c = i & 255;
    dst[m * HID + c]        = h1[i];
    dst[m * HID + HALF + c] = h2[i];
  }
}

extern "C" void kernel_launch(void* const* d_in, const int* in_sizes, int n_in,
                              void* d_out, int out_size, void* d_ws, size_t ws_size,
                              hipStream_t stream) {
  Params p;
  p.seq = (const int*)d_in[0];
  p.emb = (const float*)d_in[1];
  p.out   = (float*)d_out;
  p.flags = (int*)d_ws;
  p.ring  = (float*)((char*)d_ws + 256);

  // packed weights live in ws after the ring buffer
  float* pw = p.ring + (size_t)RING * BATCH * HID;
  const int wsizes[4] = { KDIM * HID, KDIM * HALF, KDIM * HID, KDIM * HALF };
  const int wcols[4]  = { HID, HALF, HID, HALF };
  for (int l = 0; l < 2; ++l) {
    const int base = 2 + l * 8;
    for (int m = 0; m < 4; ++m) {
      const float* Wsrc = (const float*)d_in[base + 2 * m];
      p.B[l][m]  = (const float*)d_in[base + 2 * m + 1];
      p.PW[l][m] = pw;
      const int total = (wcols[m] / 16) * KQ * 32;
      repack_kernel<<<(total + 255) / 256, 256, 0, stream>>>(Wsrc, pw, wcols[m]);
      pw += wsizes[m];
    }
  }

  init_kernel<<<(BATCH * SLICE_ + 511) / 512, 512, 0, stream>>>(p.out, p.flags);

  const size_t smem = (size_t)4 * BATCH * HALF * sizeof(float) + BATCH * sizeof(int);
  rnn_kernel<<<dim3(2), dim3(512), smem, stream>>>(p);
}